// WNOBlock_34832184770645
// MI455X (gfx1250) — compile-verified
//
#include <hip/hip_runtime.h>
#include <math.h>

#define C_CH 256

typedef float v2f __attribute__((ext_vector_type(2)));
typedef float v8f __attribute__((ext_vector_type(8)));

// db4 filter banks (match reference exactly)
__constant__ float c_dec_lo[8] = {
    -0.010597401785069032f,  0.0328830116668852f,   0.030841381835560764f,
    -0.18703481171888114f,  -0.027983769416859854f, 0.6308807679298589f,
     0.7148465705529157f,    0.2303778133088965f };
__constant__ float c_dec_hi[8] = {
    -0.2303778133088965f,    0.7148465705529157f,  -0.6308807679298589f,
    -0.027983769416859854f,  0.18703481171888114f,  0.030841381835560764f,
    -0.0328830116668852f,   -0.010597401785069032f };
__constant__ float c_rec_lo[8] = {
     0.2303778133088965f,    0.7148465705529157f,   0.6308807679298589f,
    -0.027983769416859854f, -0.18703481171888114f,  0.030841381835560764f,
     0.0328830116668852f,   -0.010597401785069032f };
__constant__ float c_rec_hi[8] = {
    -0.010597401785069032f, -0.0328830116668852f,   0.030841381835560764f,
     0.18703481171888114f,  -0.027983769416859854f,-0.6308807679298589f,
     0.7148465705529157f,   -0.2303778133088965f };

__device__ __forceinline__ float gelu_tanh(float x) {
    // jax.nn.gelu default (approximate=True)
    float x3 = x * x * x;
    return 0.5f * x * (1.0f + tanhf(0.7978845608028654f * (x + 0.044715f * x3)));
}

// ---------------------------------------------------------------------------
// Depthwise stride-2 analysis DWT: in [BC, Lin] -> approx/detail [BC, Lin/2]
// ---------------------------------------------------------------------------
__global__ void __launch_bounds__(256) dwt_kernel(
    const float* __restrict__ in, float* __restrict__ ap,
    float* __restrict__ de, int Lin)
{
    const int Lout = Lin >> 1;
    const int t  = blockIdx.x * 256 + threadIdx.x;
    const int bc = t / Lout;
    const int n  = t - bc * Lout;
    const float* p = in + (long)bc * Lin;
    float lo = 0.0f, hi = 0.0f;
    const int j0 = 2 * n - 3;
#pragma unroll
    for (int k = 0; k < 8; ++k) {
        int j = j0 + k;
        if (j >= 0 && j < Lin) {
            float v = p[j];
            lo += v * c_dec_lo[k];
            hi += v * c_dec_hi[k];
        }
    }
    ap[(long)bc * Lout + n] = lo;
    de[(long)bc * Lout + n] = hi;
}

// ---------------------------------------------------------------------------
// Transposed stride-2 synthesis DWT: approx+detail [BC, Lin] -> out [BC, 2*Lin]
// out[t] = sum_k (t+3-k even, n=(t+3-k)/2 in range)  a[n]*rec_lo[k]+d[n]*rec_hi[k]
// ---------------------------------------------------------------------------
__global__ void __launch_bounds__(256) idwt_kernel(
    const float* __restrict__ ap, const float* __restrict__ de,
    float* __restrict__ out, int Lin)
{
    const int Lout = Lin << 1;
    const int t  = blockIdx.x * 256 + threadIdx.x;
    const int bc = t / Lout;
    const int tt = t - bc * Lout;
    const float* pa = ap + (long)bc * Lin;
    const float* pd = de + (long)bc * Lin;
    float acc = 0.0f;
#pragma unroll
    for (int k = 0; k < 8; ++k) {
        int num = tt + 3 - k;
        if ((num & 1) == 0) {
            int n = num >> 1;
            if (n >= 0 && n < Lin)
                acc += pa[n] * c_rec_lo[k] + pd[n] * c_rec_hi[k];
        }
    }
    out[t] = acc;
}

// ---------------------------------------------------------------------------
// Fused ResConv1dBlock: Out = X + W2 @ gelu(W1 @ X + b1) + b2
// One 256-thread block (8 wave32) handles a 64-column stripe of one batch row.
// X tile and GELU intermediate H live in LDS; both GEMMs use
// V_WMMA_F32_16X16X4_F32. Safe to run with Out == X (in-place) because X is
// fully staged to LDS before any global write.
// ---------------------------------------------------------------------------
constexpr int NP = 68;  // padded LDS row stride (floats)

__global__ void __launch_bounds__(256) res_gemm_kernel(
    const float* __restrict__ X, float* __restrict__ Out,
    const float* __restrict__ W1, const float* __restrict__ B1,
    const float* __restrict__ W2, const float* __restrict__ B2,
    int L)
{
    __shared__ float Xs[C_CH * NP];
    __shared__ float Hs[C_CH * NP];

    const int tid  = threadIdx.x;
    const int wave = tid >> 5;
    const int lane = tid & 31;
    const int half = lane >> 4;   // lanes 16-31 hold K+2,K+3 (A/B) or M+8 (C/D)
    const int l16  = lane & 15;
    const int m0   = wave * 32;   // this wave owns output rows [m0, m0+32)

    const int stripes = L >> 6;
    const int b   = blockIdx.x / stripes;
    const int l0  = (blockIdx.x - b * stripes) << 6;
    const long xbase = (long)b * C_CH * L + l0;

    // ---- stage X[256][64] into LDS (float4, coalesced) ----
    for (int i = tid; i < C_CH * 16; i += 256) {
        int r  = i >> 4;
        int n4 = (i & 15) << 2;
        float4 v = *(const float4*)(X + xbase + (long)r * L + n4);
        *(float4*)&Xs[r * NP + n4] = v;
    }
    __syncthreads();

    const v8f zacc = {0.f, 0.f, 0.f, 0.f, 0.f, 0.f, 0.f, 0.f};
    v8f acc[2][4];

    // ================= GEMM1: H = gelu(W1 @ X + b1) =================
#pragma unroll
    for (int mt = 0; mt < 2; ++mt)
#pragma unroll
        for (int nt = 0; nt < 4; ++nt) acc[mt][nt] = zacc;

    for (int k = 0; k < C_CH; k += 4) {
        // A fragments: W1 rows m0..m0+31, cols k..k+3 (float2 per lane)
        v2f a0 = *(const v2f*)(W1 + (m0 +      l16) * C_CH + k + 2 * half);
        v2f a1 = *(const v2f*)(W1 + (m0 + 16 + l16) * C_CH + k + 2 * half);
#pragma unroll
        for (int nt = 0; nt < 4; ++nt) {
            int col = nt * 16 + l16;
            v2f bf;
            bf.x = Xs[(k +     2 * half) * NP + col];
            bf.y = Xs[(k + 1 + 2 * half) * NP + col];
            acc[0][nt] = __builtin_amdgcn_wmma_f32_16x16x4_f32(
                false, a0, false, bf, (short)0, acc[0][nt], false, false);
            acc[1][nt] = __builtin_amdgcn_wmma_f32_16x16x4_f32(
                false, a1, false, bf, (short)0, acc[1][nt], false, false);
        }
    }

    // bias + GELU, scatter H into LDS (row-major, conflict-free per half-wave)
#pragma unroll
    for (int mt = 0; mt < 2; ++mt)
#pragma unroll
        for (int nt = 0; nt < 4; ++nt)
#pragma unroll
            for (int j = 0; j < 8; ++j) {
                int row = m0 + mt * 16 + j + 8 * half;
                int col = nt * 16 + l16;
                float h = acc[mt][nt][j] + B1[row];
                Hs[row * NP + col] = gelu_tanh(h);
            }
    __syncthreads();

    // ================= GEMM2: Out = X + W2 @ H + b2 =================
#pragma unroll
    for (int mt = 0; mt < 2; ++mt)
#pragma unroll
        for (int nt = 0; nt < 4; ++nt) acc[mt][nt] = zacc;

    for (int k = 0; k < C_CH; k += 4) {
        v2f a0 = *(const v2f*)(W2 + (m0 +      l16) * C_CH + k + 2 * half);
        v2f a1 = *(const v2f*)(W2 + (m0 + 16 + l16) * C_CH + k + 2 * half);
#pragma unroll
        for (int nt = 0; nt < 4; ++nt) {
            int col = nt * 16 + l16;
            v2f bf;
            bf.x = Hs[(k +     2 * half) * NP + col];
            bf.y = Hs[(k + 1 + 2 * half) * NP + col];
            acc[0][nt] = __builtin_amdgcn_wmma_f32_16x16x4_f32(
                false, a0, false, bf, (short)0, acc[0][nt], false, false);
            acc[1][nt] = __builtin_amdgcn_wmma_f32_16x16x4_f32(
                false, a1, false, bf, (short)0, acc[1][nt], false, false);
        }
    }

    // epilogue: residual + bias, coalesced global store
#pragma unroll
    for (int mt = 0; mt < 2; ++mt)
#pragma unroll
        for (int nt = 0; nt < 4; ++nt)
#pragma unroll
            for (int j = 0; j < 8; ++j) {
                int row = m0 + mt * 16 + j + 8 * half;
                int col = nt * 16 + l16;
                float v = acc[mt][nt][j] + B2[row] + Xs[row * NP + col];
                Out[xbase + (long)row * L + col] = v;
            }
}

// ---------------------------------------------------------------------------
// LayerNorm over channels of (y + skip): one wave32 per position, 8 ch/lane.
// ---------------------------------------------------------------------------
__global__ void __launch_bounds__(256) ln_kernel(
    const float* __restrict__ y, const float* __restrict__ x,
    const float* __restrict__ g, const float* __restrict__ beta,
    float* __restrict__ out, int L)
{
    const int wave = threadIdx.x >> 5;
    const int lane = threadIdx.x & 31;
    const long p = (long)blockIdx.x * 8 + wave;   // position in [0, B*L)
    const int b = (int)(p / L);
    const int l = (int)(p - (long)b * L);
    const long base = ((long)b * C_CH) * L + l;

    float v[8];
    float s = 0.0f, s2 = 0.0f;
#pragma unroll
    for (int i = 0; i < 8; ++i) {
        int c = lane + 32 * i;
        float t = y[base + (long)c * L] + x[base + (long)c * L];
        v[i] = t; s += t; s2 += t * t;
    }
#pragma unroll
    for (int o = 16; o > 0; o >>= 1) {
        s  += __shfl_xor(s,  o, 32);
        s2 += __shfl_xor(s2, o, 32);
    }
    const float mu  = s * (1.0f / C_CH);
    const float var = s2 * (1.0f / C_CH) - mu * mu;
    const float r   = rsqrtf(var + 1e-5f);
#pragma unroll
    for (int i = 0; i < 8; ++i) {
        int c = lane + 32 * i;
        out[base + (long)c * L] = (v[i] - mu) * r * g[c] + beta[c];
    }
}

// ---------------------------------------------------------------------------
extern "C" void kernel_launch(void* const* d_in, const int* in_sizes, int n_in,
                              void* d_out, int out_size, void* d_ws, size_t ws_size,
                              hipStream_t stream) {
    (void)in_sizes; (void)n_in; (void)out_size; (void)ws_size;

    const float* x     = (const float*)d_in[0];
    const float* ab_w1 = (const float*)d_in[1];
    const float* ab_b1 = (const float*)d_in[2];
    const float* ab_w2 = (const float*)d_in[3];
    const float* ab_b2 = (const float*)d_in[4];
    const float* db_w1 = (const float*)d_in[5];
    const float* db_b1 = (const float*)d_in[6];
    const float* db_w2 = (const float*)d_in[7];
    const float* db_b2 = (const float*)d_in[8];
    const float* ln_g  = (const float*)d_in[9];
    const float* ln_b  = (const float*)d_in[10];
    float* out = (float*)d_out;
    float* ws  = (float*)d_ws;

    const long B = 16;
    const long sz1 = B * C_CH * 4096;   // 16,777,216 floats
    const long sz2 = B * C_CH * 2048;
    const long sz3 = B * C_CH * 1024;

    // workspace layout with lifetime-based aliasing (~288 MB total)
    float* d1 = ws;                              // detail L1 (res in-place)
    float* a1 = ws + sz1;                        // approx L1, later y1
    float* d2 = ws + 2 * sz1;                    // detail L2 (res in-place)
    float* a2 = ws + 2 * sz1 + sz2;              // approx L2, later y3
    float* d3 = ws + 2 * sz1 + 2 * sz2;          // detail L3 (res in-place)
    float* a3 = ws + 2 * sz1 + 2 * sz2 + sz3;    // approx L3 (res in-place)
    float* y2 = ws + 2 * sz1 + 2 * sz2 + 2 * sz3;

    // ---- analysis ----
    dwt_kernel<<<(int)(B * C_CH * 4096 / 256), 256, 0, stream>>>(x,  a1, d1, 8192);
    res_gemm_kernel<<<(int)(B * 4096 / 64), 256, 0, stream>>>(d1, d1, db_w1, db_b1, db_w2, db_b2, 4096);
    dwt_kernel<<<(int)(B * C_CH * 2048 / 256), 256, 0, stream>>>(a1, a2, d2, 4096);
    res_gemm_kernel<<<(int)(B * 2048 / 64), 256, 0, stream>>>(d2, d2, db_w1, db_b1, db_w2, db_b2, 2048);
    dwt_kernel<<<(int)(B * C_CH * 1024 / 256), 256, 0, stream>>>(a2, a3, d3, 2048);
    res_gemm_kernel<<<(int)(B * 1024 / 64), 256, 0, stream>>>(d3, d3, db_w1, db_b1, db_w2, db_b2, 1024);
    // approx branch
    res_gemm_kernel<<<(int)(B * 1024 / 64), 256, 0, stream>>>(a3, a3, ab_w1, ab_b1, ab_w2, ab_b2, 1024);

    // ---- synthesis ----
    idwt_kernel<<<(int)(B * C_CH * 2048 / 256), 256, 0, stream>>>(a3, d3, a2, 1024); // y3 -> a2 slot
    idwt_kernel<<<(int)(B * C_CH * 4096 / 256), 256, 0, stream>>>(a2, d2, y2, 2048); // y2
    idwt_kernel<<<(int)(B * C_CH * 8192 / 256), 256, 0, stream>>>(y2, d1, a1, 4096); // y1 -> a1 slot

    // ---- skip + layernorm ----
    ln_kernel<<<(int)(B * 8192 / 8), 256, 0, stream>>>(a1, x, ln_g, ln_b, out, 8192);
}